// GCN_85796266705527
// MI455X (gfx1250) — compile-verified
//
#include <hip/hip_runtime.h>
#include <hip/hip_bf16.h>

typedef float v2f __attribute__((ext_vector_type(2)));
typedef float v8f __attribute__((ext_vector_type(8)));

#define HID 32

// ---------------- small elementwise helpers ----------------

__global__ void fill_kernel(float* __restrict__ p, float v, int n) {
  int i = blockIdx.x * blockDim.x + threadIdx.x;
  if (i < n) p[i] = v;
}

// deg[col] += ew[e]   (deg pre-initialized to 1.0 = self-loop weight)
__global__ void deg_kernel(const int* __restrict__ ei, const float* __restrict__ ew,
                           float* __restrict__ deg, int E) {
  int e = blockIdx.x * blockDim.x + threadIdx.x;
  if (e < E) {
    int c = ei[E + e];          // ei[1][e]
    atomicAdd(&deg[c], ew[e]);
  }
}

__global__ void rsqrt_kernel(float* __restrict__ p, int n) {
  int i = blockIdx.x * blockDim.x + threadIdx.x;
  if (i < n) p[i] = rsqrtf(p[i]);   // deg >= 1 always (self loop), so no zero guard needed
}

// ---------------- f32 WMMA GEMM:  Y[nrows,32] = X[nrows,K] @ W[K,32] ----------------
// One wave computes one 16x16 output tile; 2 column tiles cover the 32 output cols.
// A (16x4 f32): lanes 0-15 hold K=0,1 in v[0],v[1]; lanes 16-31 hold K=2,3.
// B (4x16 f32): same K split across lane halves, N = lane%16.
// C/D (16x16 f32): VGPR j = row (j + 8*half), col = lane%16.
template <int K>
__global__ void gemm_wmma(const float* __restrict__ X, const float* __restrict__ W,
                          float* __restrict__ Y, int nrows) {
  int wave = blockIdx.x * (blockDim.x >> 5) + (threadIdx.x >> 5);
  int rowTiles = (nrows + 15) >> 4;
  int tm = wave >> 1;                 // row tile
  int tn = wave & 1;                  // col tile (0 or 1)
  if (tm >= rowTiles) return;         // wave-uniform: EXEC stays all-ones for WMMA

  int lane = threadIdx.x & 31;
  int half = lane >> 4;               // 0: K=0..1, 1: K=2..3
  int l16  = lane & 15;
  int m0 = tm << 4;
  int n0 = tn << 4;

  int mrow = m0 + l16;
  if (mrow >= nrows) mrow = nrows - 1;          // clamp (tiles are full when nrows%16==0)
  const float* xr = X + (size_t)mrow * K;
  const float* wc = W + n0 + l16;

  v8f acc = {};
#pragma unroll
  for (int k = 0; k < K; k += 4) {
    int ka = k + (half << 1);
    v2f a = *reinterpret_cast<const v2f*>(xr + ka);   // 8B-aligned: ka is even
    v2f b;
    b.x = wc[(size_t)ka * HID];
    b.y = wc[(size_t)(ka + 1) * HID];
    acc = __builtin_amdgcn_wmma_f32_16x16x4_f32(
        /*neg_a=*/false, a, /*neg_b=*/false, b,
        /*c_mod=*/(short)0, acc, /*reuse_a=*/false, /*reuse_b=*/false);
  }

#pragma unroll
  for (int j = 0; j < 8; ++j) {
    int r = m0 + (half << 3) + j;
    if (r < nrows) Y[(size_t)r * HID + n0 + l16] = acc[j];
  }
}

// dst[v,f] = dinv[v]^2 * src[v,f]   (self-loop contribution, norm = dinv*1*dinv)
__global__ void selfloop_init_kernel(const float* __restrict__ src,
                                     const float* __restrict__ dinv,
                                     float* __restrict__ dst, int n) {
  int idx = blockIdx.x * blockDim.x + threadIdx.x;
  if (idx < n * HID) {
    float d = dinv[idx >> 5];
    dst[idx] = d * d * src[idx];
  }
}

// wave-per-edge, lane-per-feature scatter:
// dst[col,f] += dinv[row]*ew*dinv[col] * src[row,f]
__global__ void scatter_kernel(const int* __restrict__ ei, const float* __restrict__ ew,
                               const float* __restrict__ dinv,
                               const float* __restrict__ src, float* __restrict__ dst,
                               int E) {
  int idx = blockIdx.x * blockDim.x + threadIdx.x;
  int e = idx >> 5;
  if (e < E) {
    int f = idx & 31;
    int r = ei[e];
    int c = ei[E + e];
    float nrm = dinv[r] * ew[e] * dinv[c];      // uniform across the wave (broadcast loads)
    atomicAdd(&dst[(size_t)c * HID + f], nrm * src[(size_t)r * HID + f]);
  }
}

// Fused: bias + ReLU + attention softmax across 3 branches + weighted combine.
// One wave per node, lane = feature. Also writes weight outputs w1,w2,w3.
__global__ void attn_kernel(const float* __restrict__ emb,   // 3 contiguous [n*32] blocks
                            const float* __restrict__ b1, const float* __restrict__ b2,
                            const float* __restrict__ b3, const float* __restrict__ att_w,
                            float* __restrict__ h, float* __restrict__ wout, int n) {
  int v = blockIdx.x * (blockDim.x >> 5) + (threadIdx.x >> 5);
  if (v >= n) return;
  int f = threadIdx.x & 31;
  size_t base = (size_t)v * HID + f;
  size_t g = (size_t)n * HID;

  float aw = att_w[f];
  float e1 = fmaxf(emb[base] + b1[f], 0.0f);
  float e2 = fmaxf(emb[g + base] + b2[f], 0.0f);
  float e3 = fmaxf(emb[2 * g + base] + b3[f], 0.0f);

  float s1 = e1 * aw, s2 = e2 * aw, s3 = e3 * aw;
#pragma unroll
  for (int off = 16; off > 0; off >>= 1) {     // wave32 tree reduction
    s1 += __shfl_xor(s1, off, 32);
    s2 += __shfl_xor(s2, off, 32);
    s3 += __shfl_xor(s3, off, 32);
  }
  float c1 = expf(s1 > 0.0f ? s1 : 0.01f * s1);   // leaky_relu slope 0.01, then exp
  float c2 = expf(s2 > 0.0f ? s2 : 0.01f * s2);
  float c3 = expf(s3 > 0.0f ? s3 : 0.01f * s3);
  float inv = 1.0f / (c1 + c2 + c3);
  float w1 = c1 * inv, w2 = c2 * inv, w3 = c3 * inv;

  h[base] = w1 * e1 + w2 * e2 + w3 * e3;
  if (f == 0) {
    wout[v]         = w1;
    wout[n + v]     = w2;
    wout[2 * n + v] = w3;
  }
}

// out[v,f] = sum_i bo_i[f] + sum_i dinv_i[v]^2 * hw_i[v,f]
__global__ void out_init_kernel(const float* __restrict__ hw, const float* __restrict__ dinv,
                                const float* __restrict__ bo1, const float* __restrict__ bo2,
                                const float* __restrict__ bo3, float* __restrict__ out, int n) {
  int idx = blockIdx.x * blockDim.x + threadIdx.x;
  if (idx >= n * HID) return;
  int v = idx >> 5;
  int f = idx & 31;
  size_t g = (size_t)n * HID;
  float d1 = dinv[v], d2 = dinv[n + v], d3 = dinv[2 * n + v];
  out[idx] = bo1[f] + bo2[f] + bo3[f]
           + d1 * d1 * hw[idx] + d2 * d2 * hw[g + idx] + d3 * d3 * hw[2 * g + idx];
}

// ---------------- host orchestration ----------------

extern "C" void kernel_launch(void* const* d_in, const int* in_sizes, int n_in,
                              void* d_out, int out_size, void* d_ws, size_t ws_size,
                              hipStream_t stream) {
  const int n = in_sizes[0] / 128;        // N nodes (x1 is [N,128])

  // inputs: per branch i: [7i]=x, [7i+1]=ei, [7i+2]=ea, [7i+3]=W, [7i+4]=b, [7i+5]=Wo, [7i+6]=bo
  const float* x[3]; const int* ei[3]; const float* ea[3];
  const float* W[3]; const float* b[3]; const float* Wo[3]; const float* bo[3];
  int E[3];
  for (int i = 0; i < 3; ++i) {
    x[i]  = (const float*)d_in[7 * i + 0];
    ei[i] = (const int*)  d_in[7 * i + 1];
    ea[i] = (const float*)d_in[7 * i + 2];
    W[i]  = (const float*)d_in[7 * i + 3];
    b[i]  = (const float*)d_in[7 * i + 4];
    Wo[i] = (const float*)d_in[7 * i + 5];
    bo[i] = (const float*)d_in[7 * i + 6];
    E[i]  = in_sizes[7 * i + 1] / 2;      // ei is [2,E]
  }
  const float* att_w = (const float*)d_in[21];

  // workspace layout (floats): dinv[3n] | xw/hw[3*n*32] | emb[3*n*32] | h[n*32]
  float* ws   = (float*)d_ws;
  float* dinv = ws;                               // 3n (deg, then rsqrt in place)
  float* xw   = dinv + 3 * (size_t)n;             // 96n (reused for hw in layer 2)
  float* emb  = xw + 96 * (size_t)n;              // 96n
  float* h    = emb + 96 * (size_t)n;             // 32n
  float* out  = (float*)d_out;                    // n*32 out, then 3*n weights

  const int T = 256;
  int nf = n * HID;

  // 1) degrees (self loop = 1.0), then dinv = rsqrt(deg)
  fill_kernel<<<(3 * n + T - 1) / T, T, 0, stream>>>(dinv, 1.0f, 3 * n);
  for (int i = 0; i < 3; ++i)
    deg_kernel<<<(E[i] + T - 1) / T, T, 0, stream>>>(ei[i], ea[i], dinv + (size_t)i * n, E[i]);
  rsqrt_kernel<<<(3 * n + T - 1) / T, T, 0, stream>>>(dinv, 3 * n);

  // 2) xw_i = x_i @ W_i  (WMMA, K=128)
  int rowTiles = (n + 15) / 16;
  int gemmBlocks = (rowTiles * 2 + 3) / 4;        // 4 waves per 128-thread block
  for (int i = 0; i < 3; ++i)
    gemm_wmma<128><<<gemmBlocks, 128, 0, stream>>>(x[i], W[i], xw + (size_t)i * nf, n);

  // 3) propagate layer 1: self loops + edge scatter into emb_i
  for (int i = 0; i < 3; ++i)
    selfloop_init_kernel<<<(nf + T - 1) / T, T, 0, stream>>>(
        xw + (size_t)i * nf, dinv + (size_t)i * n, emb + (size_t)i * nf, n);
  for (int i = 0; i < 3; ++i)
    scatter_kernel<<<(E[i] * 32 + T - 1) / T, T, 0, stream>>>(
        ei[i], ea[i], dinv + (size_t)i * n, xw + (size_t)i * nf, emb + (size_t)i * nf, E[i]);

  // 4) bias + ReLU + attention + combine -> h; weights -> d_out tail
  attn_kernel<<<(n + 7) / 8, 256, 0, stream>>>(emb, b[0], b[1], b[2], att_w,
                                               h, out + (size_t)n * HID, n);

  // 5) hw_i = h @ Wo_i  (WMMA, K=32), reuse xw buffer
  for (int i = 0; i < 3; ++i)
    gemm_wmma<32><<<gemmBlocks, 128, 0, stream>>>(h, Wo[i], xw + (size_t)i * nf, n);

  // 6) layer 2: self loops + Σ biases, then 3 atomic scatters straight into d_out
  out_init_kernel<<<(nf + T - 1) / T, T, 0, stream>>>(xw, dinv, bo[0], bo[1], bo[2], out, n);
  for (int i = 0; i < 3; ++i)
    scatter_kernel<<<(E[i] * 32 + T - 1) / T, T, 0, stream>>>(
        ei[i], ea[i], dinv + (size_t)i * n, xw + (size_t)i * nf, out, E[i]);
}